// EntityAggregator_63342177681721
// MI455X (gfx1250) — compile-verified
//
#include <hip/hip_runtime.h>

// ---------------------------------------------------------------------------
// EntityAggregator for MI455X (gfx1250).
// Roofline: W_r is 268MB read once (~11.5us @ 23.3TB/s) -> entity_side_kernel
// is a pure HBM stream (float4 loads + prefetch). The three genuine GEMMs
// (item@W_ui^T, (item+ego)@linUI_W^T, agg@lin_W^T) use V_WMMA_F32_16X16X4_F32
// so the matrix pipe is used where matmul structure actually exists, in fp32.
// ---------------------------------------------------------------------------

#define B_   128
#define N_   8
#define S_   16
#define DIM_ 64
#define H_   4
#define DH_  16

typedef float v2f __attribute__((ext_vector_type(2)));
typedef float v8f __attribute__((ext_vector_type(8)));

// D(16x16) += A(16xK) * W^T(Kx16), A row-major [16 x lda], W row-major
// [16 x ldw] slice (we multiply by W^T, so B[k,n] = W[n,k]).
// WMMA f32 16x16x4 fragment layout (ISA 7.12.2):
//   A: lane<16 -> {A[l,k], A[l,k+1]}, lane>=16 -> {A[l,k+2], A[l,k+3]}
//   B: lane<16 -> {B[k,l], B[k+1,l]}, lane>=16 -> {B[k+2,l], B[k+3,l]}
//   D: vgpr r  -> lane<16: (M=r, N=l); lane>=16: (M=r+8, N=l)
__device__ inline v8f wmma_tile_AxWt(const float* A, int lda,
                                     const float* W, int ldw,
                                     int K, v8f c) {
  const int lane = threadIdx.x & 31;
  const int half = lane >> 4;
  const int l    = lane & 15;
  for (int k = 0; k < K; k += 4) {
    const int kk = k + half * 2;
    v2f a, b;
    a.x = A[l * lda + kk];
    a.y = A[l * lda + kk + 1];
    b.x = W[l * ldw + kk];      // B[kk,   n=l] = W[l, kk]
    b.y = W[l * ldw + kk + 1];  // B[kk+1, n=l]
    c = __builtin_amdgcn_wmma_f32_16x16x4_f32(false, a, false, b,
                                              (short)0, c, false, false);
  }
  return c;
}

// ---------------------------------------------------------------------------
// Kernel 1: user-side. 8 blocks x 256 threads; block handles 16 batch rows.
//   Wi = item @ W_ui^T (WMMA) -> per-(b,h) attention over S -> user_ego
//   signal = user + relu((item + user_ego) @ linUI_W^T + b)   (WMMA)
// ---------------------------------------------------------------------------
__global__ __launch_bounds__(256) void user_side_kernel(
    const float* __restrict__ item,      // [B,64]
    const float* __restrict__ user,      // [B,64]
    const float* __restrict__ nghu,      // [B,S,64]
    const float* __restrict__ W_ui,      // [64,64]
    const float* __restrict__ linUI_W,   // [64,64]
    const float* __restrict__ linUI_b,   // [64]
    float* __restrict__ user_ego,        // [B,64] (ws)
    float* __restrict__ signal)          // [B,64] (ws)
{
  __shared__ float sWi[16][DIM_];
  __shared__ float sX[16][DIM_];
  const int m0   = blockIdx.x * 16;
  const int tid  = threadIdx.x;
  const int wave = tid >> 5;
  const int lane = tid & 31;
  const int half = lane >> 4;
  const int l    = lane & 15;

  // ---- GEMM1: Wi tile (16x64) via 4 WMMA waves -------------------------
  if (wave < 4) {
    const int n0 = wave * 16;
    v8f c = {0.f, 0.f, 0.f, 0.f, 0.f, 0.f, 0.f, 0.f};
    c = wmma_tile_AxWt(item + (size_t)m0 * DIM_, DIM_,
                       W_ui + (size_t)n0 * DIM_, DIM_, DIM_, c);
#pragma unroll
    for (int r = 0; r < 8; ++r)
      sWi[r + half * 8][n0 + l] = c[r];
  }
  __syncthreads();

  // ---- attention over neighbors: one thread per (batch-in-tile, head) --
  if (tid < 64) {
    const int bl = tid >> 2, h = tid & 3;
    const int b  = m0 + bl;
    const float* nb = nghu + (size_t)b * S_ * DIM_ + h * DH_;
    float att[S_];
    float mx = -3.4e38f;
#pragma unroll
    for (int s = 0; s < S_; ++s) {
      float acc = 0.f;
#pragma unroll
      for (int d = 0; d < DH_; ++d)
        acc += sWi[bl][h * DH_ + d] * nb[s * DIM_ + d];
      acc *= 0.25f;                 // 1/sqrt(DH)
      att[s] = acc;
      mx = fmaxf(mx, acc);
    }
    float sum = 0.f;
#pragma unroll
    for (int s = 0; s < S_; ++s) { att[s] = __expf(att[s] - mx); sum += att[s]; }
    const float inv = 1.f / sum;
#pragma unroll
    for (int d = 0; d < DH_; ++d) {
      float e = 0.f;
#pragma unroll
      for (int s = 0; s < S_; ++s) e += att[s] * nb[s * DIM_ + d];
      e *= inv;
      user_ego[(size_t)b * DIM_ + h * DH_ + d] = e;
      sX[bl][h * DH_ + d] = item[(size_t)b * DIM_ + h * DH_ + d] + e;
    }
  }
  __syncthreads();

  // ---- GEMM2: signal = user + relu(sX @ linUI_W^T + b) -----------------
  if (wave < 4) {
    const int n0 = wave * 16;
    v8f c = {0.f, 0.f, 0.f, 0.f, 0.f, 0.f, 0.f, 0.f};
    c = wmma_tile_AxWt(&sX[0][0], DIM_,
                       linUI_W + (size_t)n0 * DIM_, DIM_, DIM_, c);
#pragma unroll
    for (int r = 0; r < 8; ++r) {
      const int M   = m0 + r + half * 8;
      const int col = n0 + l;
      const float iu = fmaxf(c[r] + linUI_b[col], 0.f);
      signal[(size_t)M * DIM_ + col] = user[(size_t)M * DIM_ + col] + iu;
    }
  }
}

// ---------------------------------------------------------------------------
// Kernel 2: entity-side, one block per (b,n). Pure HBM stream of the 256KB
// W_r tile (this kernel carries ~100% of the bytes -> the whole problem).
// ---------------------------------------------------------------------------
__global__ __launch_bounds__(256) void entity_side_kernel(
    const float* __restrict__ selfE,    // [B,N,64]
    const float* __restrict__ nghe,     // [B,N,S,64]
    const float* __restrict__ W_r,      // [B,N,S,64,64]
    const float* __restrict__ signal,   // [B,64]
    const float* __restrict__ user_ego, // [B,64]
    float* __restrict__ agg)            // [B,N,64] (ws)
{
  __shared__ float v[DIM_];
  __shared__ float q[S_][DIM_];
  __shared__ float att[H_][S_];
  const int bn  = blockIdx.x;       // b*N + n
  const int b   = bn >> 3;
  const int tid = threadIdx.x;
  const float* selfp = selfE + (size_t)bn * DIM_;

  if (tid < DIM_) v[tid] = signal[(size_t)b * DIM_ + tid] * selfp[tid];
  __syncthreads();

  // q[s][i] = W_r[b,n,s,i,:] . v   — 1024 rows, 4 per thread, float4 stream
  const float* Wb = W_r + (size_t)bn * S_ * DIM_ * DIM_;
#pragma unroll
  for (int rr = 0; rr < 4; ++rr) {
    const int row = tid + rr * 256;
    if (rr < 3)  // emits global_prefetch_b8 for the next 256KB/4 chunk
      __builtin_prefetch(Wb + (size_t)(tid + (rr + 1) * 256) * DIM_, 0, 0);
    const float4* wrow = reinterpret_cast<const float4*>(Wb + (size_t)row * DIM_);
    float acc = 0.f;
#pragma unroll
    for (int j4 = 0; j4 < 16; ++j4) {
      const float4 w = wrow[j4];
      acc += w.x * v[j4 * 4 + 0] + w.y * v[j4 * 4 + 1] +
             w.z * v[j4 * 4 + 2] + w.w * v[j4 * 4 + 3];
    }
    q[row >> 6][row & 63] = acc;
  }
  __syncthreads();

  // att_e raw scores: one thread per (h,s)
  const float* ne = nghe + (size_t)bn * S_ * DIM_;
  if (tid < 64) {
    const int h = tid >> 4, s = tid & 15;
    float acc = 0.f;
#pragma unroll
    for (int d = 0; d < DH_; ++d)
      acc += q[s][h * DH_ + d] * ne[s * DIM_ + h * DH_ + d];
    att[h][s] = acc * 0.25f;
  }
  __syncthreads();

  // softmax over S per head
  if (tid < 4) {
    float mx = -3.4e38f;
#pragma unroll
    for (int s = 0; s < S_; ++s) mx = fmaxf(mx, att[tid][s]);
    float tmp[S_];
    float sum = 0.f;
#pragma unroll
    for (int s = 0; s < S_; ++s) { tmp[s] = __expf(att[tid][s] - mx); sum += tmp[s]; }
    const float inv = 1.f / sum;
#pragma unroll
    for (int s = 0; s < S_; ++s) att[tid][s] = tmp[s] * inv;
  }
  __syncthreads();

  // entity_ego + final pre-GEMM aggregation
  if (tid < 64) {
    const int h = tid >> 4;
    float ego = 0.f;
#pragma unroll
    for (int s = 0; s < S_; ++s) ego += att[h][s] * ne[s * DIM_ + tid];
    agg[(size_t)bn * DIM_ + tid] =
        selfp[tid] + user_ego[(size_t)b * DIM_ + tid] + ego;
  }
}

// ---------------------------------------------------------------------------
// Kernel 3: out = relu(agg @ lin_W^T + lin_b), M=1024, N=K=64. 256 WMMA tiles
// across 32 blocks x 8 waves.
// ---------------------------------------------------------------------------
__global__ __launch_bounds__(256) void out_gemm_kernel(
    const float* __restrict__ agg,    // [1024,64]
    const float* __restrict__ lin_W,  // [64,64]
    const float* __restrict__ lin_b,  // [64]
    float* __restrict__ out)          // [1024,64]
{
  const int wave = threadIdx.x >> 5;
  const int tile = blockIdx.x * 8 + wave;     // 0..255
  const int m0 = (tile >> 2) * 16;
  const int n0 = (tile & 3) * 16;
  v8f c = {0.f, 0.f, 0.f, 0.f, 0.f, 0.f, 0.f, 0.f};
  c = wmma_tile_AxWt(agg + (size_t)m0 * DIM_, DIM_,
                     lin_W + (size_t)n0 * DIM_, DIM_, DIM_, c);
  const int lane = threadIdx.x & 31;
  const int half = lane >> 4;
  const int l    = lane & 15;
#pragma unroll
  for (int r = 0; r < 8; ++r) {
    const int M   = m0 + r + half * 8;
    const int col = n0 + l;
    out[(size_t)M * DIM_ + col] = fmaxf(c[r] + lin_b[col], 0.f);
  }
}

// ---------------------------------------------------------------------------
extern "C" void kernel_launch(void* const* d_in, const int* in_sizes, int n_in,
                              void* d_out, int out_size, void* d_ws, size_t ws_size,
                              hipStream_t stream) {
  const float* selfE   = (const float*)d_in[0];   // [B,N,DIM]
  const float* nghu    = (const float*)d_in[1];   // [B,S,DIM]
  const float* nghe    = (const float*)d_in[2];   // [B,N,S,DIM]
  const float* user    = (const float*)d_in[3];   // [B,DIM]
  const float* item    = (const float*)d_in[4];   // [B,DIM]
  const float* W_r     = (const float*)d_in[5];   // [B,N,S,DIM,DIM]
  const float* W_ui    = (const float*)d_in[6];   // [DIM,DIM]
  const float* lin_W   = (const float*)d_in[7];   // [DIM,DIM]
  const float* lin_b   = (const float*)d_in[8];   // [DIM]
  const float* linUI_W = (const float*)d_in[9];   // [DIM,DIM]
  const float* linUI_b = (const float*)d_in[10];  // [DIM]
  // d_in[11] = is_item_layer (==1); reference code path is unconditional.

  float* ws       = (float*)d_ws;
  float* user_ego = ws;                 // B*DIM   = 8192 floats
  float* signal   = ws + B_ * DIM_;     // B*DIM   = 8192 floats
  float* agg      = ws + 2 * B_ * DIM_; // B*N*DIM = 65536 floats

  user_side_kernel<<<B_ / 16, 256, 0, stream>>>(
      item, user, nghu, W_ui, linUI_W, linUI_b, user_ego, signal);
  entity_side_kernel<<<B_ * N_, 256, 0, stream>>>(
      selfE, nghe, W_r, signal, user_ego, agg);
  out_gemm_kernel<<<(B_ * N_ * 4) / (8 * 16), 256, 0, stream>>>(
      agg, lin_W, lin_b, (float*)d_out);
}